// Atom_Gloal_37958920962359
// MI455X (gfx1250) — compile-verified
//
#include <hip/hip_runtime.h>
#include <math.h>

typedef __attribute__((ext_vector_type(2))) float v2f;
typedef __attribute__((ext_vector_type(8))) float v8f;

#define N_NODES  200000
#define N_EDGES  800000
#define N_GRAPHS 8192
#define DIM_DRUG 128
#define DIM_CELL 954
#define HID      128
#define OUTF     256
#define BN_EPS   1e-5f

// ---------------- utility kernels ----------------

__global__ void fill_kernel(float* __restrict__ p, float v, long n) {
  long i = (long)blockIdx.x * blockDim.x + threadIdx.x;
  if (i < n) p[i] = v;
}

__global__ void deg_count_kernel(const int* __restrict__ dst, float* __restrict__ deg, int E) {
  int e = blockIdx.x * blockDim.x + threadIdx.x;
  if (e < E) atomicAdd(&deg[dst[e]], 1.0f);
}

// deg -> dinv = rsqrt(deg + 1)   (+1 = self loop)
__global__ void dinv_kernel(float* __restrict__ deg, int n) {
  int i = blockIdx.x * blockDim.x + threadIdx.x;
  if (i < n) deg[i] = rsqrtf(deg[i] + 1.0f);
}

// ---------------- WMMA fp32 GEMM: Y[N,F] = X[N,K] @ W[K,F] (+epilogue) ------
// One wave computes a 16x16 tile with V_WMMA_F32_16X16X4_F32.
// Compile-time K/F -> no per-iteration guards; A loaded as b64; two
// independent accumulator chains for WMMA ILP.
// MODE: 0 = plain store, 1 = tanh(acc+bias), 2 = relu(acc+bias)
template<int K, int F, int MODE>
__global__ void gemm_wmma_kernel(const float* __restrict__ X,
                                 const float* __restrict__ W,
                                 const float* __restrict__ bias,
                                 float* __restrict__ Y) {
  const int lane = threadIdx.x & 31;
  const int wave = threadIdx.x >> 5;
  const int m0   = blockIdx.x << 4;
  const int n0   = wave << 4;
  const int m    = lane & 15;
  const int half = lane >> 4;          // 0: K pair {k,k+1}, 1: {k+2,k+3}
  const int col  = n0 + (lane & 15);

  const float* __restrict__ xr0 = X + (long)(m0 + m) * K;   // row base
  const float* __restrict__ xr  = xr0 + 2 * half;
  const float* __restrict__ wc  = W + (long)(2 * half) * F + col;

  v8f acc0 = {}, acc1 = {};

  constexpr int K8 = K & ~7;
  constexpr int K4 = K & ~3;

#pragma unroll 4
  for (int k = 0; k < K8; k += 8) {
    v2f a0 = *(const v2f*)(xr + k);
    v2f b0;
    b0.x = wc[(long)k * F];
    b0.y = wc[(long)(k + 1) * F];
    acc0 = __builtin_amdgcn_wmma_f32_16x16x4_f32(false, a0, false, b0,
                                                 (short)0, acc0, false, false);
    v2f a1 = *(const v2f*)(xr + k + 4);
    v2f b1;
    b1.x = wc[(long)(k + 4) * F];
    b1.y = wc[(long)(k + 5) * F];
    acc1 = __builtin_amdgcn_wmma_f32_16x16x4_f32(false, a1, false, b1,
                                                 (short)0, acc1, false, false);
  }
  if constexpr (K8 < K4) {             // one more unguarded 4-step
    v2f a = *(const v2f*)(xr + K8);
    v2f b;
    b.x = wc[(long)K8 * F];
    b.y = wc[(long)(K8 + 1) * F];
    acc0 = __builtin_amdgcn_wmma_f32_16x16x4_f32(false, a, false, b,
                                                 (short)0, acc0, false, false);
  }
  if constexpr (K4 < K) {              // guarded tail (only K=954 path)
    const int ka = K4 + 2 * half;
    v2f a = {0.0f, 0.0f};
    v2f b = {0.0f, 0.0f};
    if (ka < K)     { a.x = xr0[ka];     b.x = W[(long)ka * F + col]; }
    if (ka + 1 < K) { a.y = xr0[ka + 1]; b.y = W[(long)(ka + 1) * F + col]; }
    acc1 = __builtin_amdgcn_wmma_f32_16x16x4_f32(false, a, false, b,
                                                 (short)0, acc1, false, false);
  }

  float bval = (MODE != 0) ? bias[col] : 0.0f;
#pragma unroll
  for (int v = 0; v < 8; ++v) {
    float val = acc0[v] + acc1[v];
    if constexpr (MODE == 1) val = tanhf(val + bval);
    if constexpr (MODE == 2) val = fmaxf(val + bval, 0.0f);
    Y[(long)(m0 + v + 8 * half) * F + col] = val;
  }
}

// ---------------- GCN aggregation (float4 paths) ----------------

// out[i,c] = xw[i,c]*dinv[i]^2 + bias[c]   (self-loop term + bias)
template<int F>
__global__ void agg_init_kernel(const float4* __restrict__ xw,
                                const float* __restrict__ dinv,
                                const float* __restrict__ bias,
                                float4* __restrict__ out, long n4) {
  long i = (long)blockIdx.x * blockDim.x + threadIdx.x;
  if (i >= n4) return;
  constexpr int G = F / 4;
  long row = i / G;
  int  c   = (int)(i - row * G) * 4;
  float di = dinv[row];
  float s  = di * di;
  float4 x  = xw[i];
  float4 bb = *(const float4*)(bias + c);
  out[i] = make_float4(x.x * s + bb.x, x.y * s + bb.y,
                       x.z * s + bb.z, x.w * s + bb.w);
}

// out[dst] += xw[src] * dinv[src]*dinv[dst]
template<int F>
__global__ void agg_edge_kernel(const float* __restrict__ xw,
                                const int* __restrict__ src,
                                const int* __restrict__ dst,
                                const float* __restrict__ dinv,
                                float* __restrict__ out, long total4) {
  long i = (long)blockIdx.x * blockDim.x + threadIdx.x;
  if (i >= total4) return;
  constexpr int G = F / 4;
  long e  = i / G;
  int  c4 = (int)(i - e * G) * 4;
  int  s  = src[e], d = dst[e];
  float nv = dinv[s] * dinv[d];
  float4 x = *(const float4*)(xw + (long)s * F + c4);
  float* o = out + (long)d * F + c4;
  atomicAdd(o + 0, x.x * nv);
  atomicAdd(o + 1, x.y * nv);
  atomicAdd(o + 2, x.z * nv);
  atomicAdd(o + 3, x.w * nv);
}

// ---------------- BatchNorm (training-mode batch stats) ----------------

// stats[0..F): sum, stats[F..2F): sumsq  (of relu(x) if relu!=0). Block=256.
template<int F>
__global__ void bn_stats_kernel(const float* __restrict__ x, float* __restrict__ stats,
                                int N, int relu, int rowsPerBlock) {
  constexpr int G    = F / 4;       // float4 column groups
  constexpr int NSUB = 256 / G;     // row subsets per block
  __shared__ float4 shs[256];
  __shared__ float4 shq[256];
  const int g    = threadIdx.x % G;
  const int rsub = threadIdx.x / G;
  const long r0  = (long)blockIdx.x * rowsPerBlock;
  float4 s = make_float4(0.f, 0.f, 0.f, 0.f);
  float4 q = make_float4(0.f, 0.f, 0.f, 0.f);
  for (int r = rsub; r < rowsPerBlock; r += NSUB) {
    long row = r0 + r;
    if (row < N) {
      float4 v = *(const float4*)(x + row * (long)F + g * 4);
      if (relu) {
        v.x = fmaxf(v.x, 0.f); v.y = fmaxf(v.y, 0.f);
        v.z = fmaxf(v.z, 0.f); v.w = fmaxf(v.w, 0.f);
      }
      s.x += v.x; s.y += v.y; s.z += v.z; s.w += v.w;
      q.x += v.x * v.x; q.y += v.y * v.y; q.z += v.z * v.z; q.w += v.w * v.w;
    }
  }
  shs[threadIdx.x] = s;
  shq[threadIdx.x] = q;
  __syncthreads();
  if (threadIdx.x < G) {
    for (int j = 1; j < NSUB; ++j) {
      float4 t = shs[threadIdx.x + j * G];
      s.x += t.x; s.y += t.y; s.z += t.z; s.w += t.w;
      t = shq[threadIdx.x + j * G];
      q.x += t.x; q.y += t.y; q.z += t.z; q.w += t.w;
    }
    const int c = g * 4;
    atomicAdd(&stats[c + 0], s.x); atomicAdd(&stats[c + 1], s.y);
    atomicAdd(&stats[c + 2], s.z); atomicAdd(&stats[c + 3], s.w);
    atomicAdd(&stats[F + c + 0], q.x); atomicAdd(&stats[F + c + 1], q.y);
    atomicAdd(&stats[F + c + 2], q.z); atomicAdd(&stats[F + c + 3], q.w);
  }
}

// stats -> stats[c]=mean, stats[F+c]=rstd
__global__ void bn_finalize_kernel(float* __restrict__ stats, int N, int F) {
  int c = threadIdx.x;
  if (c >= F) return;
  float mean = stats[c] / (float)N;
  float var  = stats[F + c] / (float)N - mean * mean;
  stats[c]     = mean;
  stats[F + c] = rsqrtf(var + BN_EPS);
}

// x = (relu?(x) - mean)*rstd*gamma + beta  (in place, float4)
template<int F>
__global__ void bn_apply_kernel(float4* __restrict__ x, const float* __restrict__ stats,
                                const float* __restrict__ gamma, const float* __restrict__ beta,
                                long n4, int relu) {
  long i = (long)blockIdx.x * blockDim.x + threadIdx.x;
  if (i >= n4) return;
  constexpr int G = F / 4;
  int c = (int)(i % G) * 4;
  float4 v  = x[i];
  if (relu) {
    v.x = fmaxf(v.x, 0.f); v.y = fmaxf(v.y, 0.f);
    v.z = fmaxf(v.z, 0.f); v.w = fmaxf(v.w, 0.f);
  }
  float4 mu = *(const float4*)(stats + c);
  float4 rs = *(const float4*)(stats + F + c);
  float4 ga = *(const float4*)(gamma + c);
  float4 be = *(const float4*)(beta + c);
  x[i] = make_float4((v.x - mu.x) * rs.x * ga.x + be.x,
                     (v.y - mu.y) * rs.y * ga.y + be.y,
                     (v.z - mu.z) * rs.z * ga.z + be.z,
                     (v.w - mu.w) * rs.w * ga.w + be.w);
}

// ---------------- segment max pool ----------------

__device__ inline void atomicMaxF(float* addr, float val) {
  if (val >= 0.0f) atomicMax((int*)addr, __float_as_int(val));
  else             atomicMin((unsigned int*)addr, __float_as_uint(val));
}

__global__ void segmax_kernel(const float* __restrict__ x, const int* __restrict__ ibatch,
                              float* __restrict__ out, long n4) {
  long i = (long)blockIdx.x * blockDim.x + threadIdx.x;
  if (i >= n4) return;
  constexpr int G = OUTF / 4;
  long row = i / G;
  int  c4  = (int)(i - row * G) * 4;
  float4 v = *(const float4*)(x + row * (long)OUTF + c4);
  float* o = out + (long)ibatch[row] * OUTF + c4;
  atomicMaxF(o + 0, v.x);
  atomicMaxF(o + 1, v.y);
  atomicMaxF(o + 2, v.z);
  atomicMaxF(o + 3, v.w);
}

// ---------------- launch ----------------

static inline int cdivl(long a, long b) { return (int)((a + b - 1) / b); }

extern "C" void kernel_launch(void* const* d_in, const int* in_sizes, int n_in,
                              void* d_out, int out_size, void* d_ws, size_t ws_size,
                              hipStream_t stream) {
  const float* drug_feature = (const float*)d_in[0];
  const int*   adj          = (const int*)d_in[1];
  const int*   esrc         = adj;
  const int*   edst         = adj + N_EDGES;
  const int*   ibatch       = (const int*)d_in[2];
  const float* gexpr        = (const float*)d_in[3];
  const float* W1  = (const float*)d_in[4];   const float* b1   = (const float*)d_in[5];
  const float* g1  = (const float*)d_in[6];   const float* be1  = (const float*)d_in[7];
  const float* W2  = (const float*)d_in[8];   const float* b2   = (const float*)d_in[9];
  const float* g2  = (const float*)d_in[10];  const float* be2  = (const float*)d_in[11];
  const float* Wc1 = (const float*)d_in[12];  const float* bc1  = (const float*)d_in[13];
  const float* gc1 = (const float*)d_in[14];  const float* bec1 = (const float*)d_in[15];
  const float* Wc2 = (const float*)d_in[16];  const float* bc2  = (const float*)d_in[17];

  // workspace layout (floats)
  float* ws    = (float*)d_ws;
  float* bufP  = ws;                               // xw2:  N*256
  float* bufQ  = ws + (long)N_NODES * OUTF;        // out2: N*256 (xw1 & agg1 live here earlier)
  float* xw1   = bufQ;                             // N*128  (dead before out2 written)
  float* agg1  = bufQ + (long)N_NODES * HID;       // N*128  (dead after GEMM2)
  float* dinv  = ws + 2L * N_NODES * OUTF;         // N
  float* c1    = dinv + N_NODES;                   // 8192*128
  float* stats = c1 + (long)N_GRAPHS * HID;        // 512

  float* xdrug = (float*)d_out;
  float* xcell = (float*)d_out + (long)N_GRAPHS * OUTF;

  const int T = 256;

  // 1) symmetric normalization factors
  fill_kernel<<<cdivl(N_NODES, T), T, 0, stream>>>(dinv, 0.0f, (long)N_NODES);
  deg_count_kernel<<<cdivl(N_EDGES, T), T, 0, stream>>>(edst, dinv, N_EDGES);
  dinv_kernel<<<cdivl(N_NODES, T), T, 0, stream>>>(dinv, N_NODES);

  // 2) GCN layer 1: xw1 = X @ W1 ; aggregate ; BN(relu) in-place
  gemm_wmma_kernel<DIM_DRUG, HID, 0><<<N_NODES / 16, 32 * (HID / 16), 0, stream>>>(
      drug_feature, W1, b1, xw1);
  agg_init_kernel<HID><<<cdivl((long)N_NODES * HID / 4, T), T, 0, stream>>>(
      (const float4*)xw1, dinv, b1, (float4*)agg1, (long)N_NODES * HID / 4);
  agg_edge_kernel<HID><<<cdivl((long)N_EDGES * HID / 4, T), T, 0, stream>>>(
      xw1, esrc, edst, dinv, agg1, (long)N_EDGES * HID / 4);
  fill_kernel<<<1, 2 * HID, 0, stream>>>(stats, 0.0f, 2L * HID);
  bn_stats_kernel<HID><<<cdivl(N_NODES, 64), T, 0, stream>>>(agg1, stats, N_NODES, 1, 64);
  bn_finalize_kernel<<<1, HID, 0, stream>>>(stats, N_NODES, HID);
  bn_apply_kernel<HID><<<cdivl((long)N_NODES * HID / 4, T), T, 0, stream>>>(
      (float4*)agg1, stats, g1, be1, (long)N_NODES * HID / 4, 1);

  // 3) GCN layer 2: xw2 = h1 @ W2 ; aggregate into bufQ ; BN(relu) in-place
  gemm_wmma_kernel<HID, OUTF, 0><<<N_NODES / 16, 32 * (OUTF / 16), 0, stream>>>(
      agg1, W2, b2, bufP);
  agg_init_kernel<OUTF><<<cdivl((long)N_NODES * OUTF / 4, T), T, 0, stream>>>(
      (const float4*)bufP, dinv, b2, (float4*)bufQ, (long)N_NODES * OUTF / 4);
  agg_edge_kernel<OUTF><<<cdivl((long)N_EDGES * OUTF / 4, T), T, 0, stream>>>(
      bufP, esrc, edst, dinv, bufQ, (long)N_EDGES * OUTF / 4);
  fill_kernel<<<1, 2 * OUTF, 0, stream>>>(stats, 0.0f, 2L * OUTF);
  bn_stats_kernel<OUTF><<<cdivl(N_NODES, 64), T, 0, stream>>>(bufQ, stats, N_NODES, 1, 64);
  bn_finalize_kernel<<<1, OUTF, 0, stream>>>(stats, N_NODES, OUTF);
  bn_apply_kernel<OUTF><<<cdivl((long)N_NODES * OUTF / 4, T), T, 0, stream>>>(
      (float4*)bufQ, stats, g2, be2, (long)N_NODES * OUTF / 4, 1);

  // 4) global max pool -> x_drug  (init to -inf = segment_max identity)
  fill_kernel<<<cdivl((long)N_GRAPHS * OUTF, T), T, 0, stream>>>(
      xdrug, -INFINITY, (long)N_GRAPHS * OUTF);
  segmax_kernel<<<cdivl((long)N_NODES * OUTF / 4, T), T, 0, stream>>>(
      bufQ, ibatch, xdrug, (long)N_NODES * OUTF / 4);

  // 5) cell branch: c1 = tanh(gexpr @ Wc1 + bc1) ; BN ; x_cell = relu(c1 @ Wc2 + bc2)
  gemm_wmma_kernel<DIM_CELL, HID, 1><<<N_GRAPHS / 16, 32 * (HID / 16), 0, stream>>>(
      gexpr, Wc1, bc1, c1);
  fill_kernel<<<1, 2 * HID, 0, stream>>>(stats, 0.0f, 2L * HID);
  bn_stats_kernel<HID><<<cdivl(N_GRAPHS, 64), T, 0, stream>>>(c1, stats, N_GRAPHS, 0, 64);
  bn_finalize_kernel<<<1, HID, 0, stream>>>(stats, N_GRAPHS, HID);
  bn_apply_kernel<HID><<<cdivl((long)N_GRAPHS * HID / 4, T), T, 0, stream>>>(
      (float4*)c1, stats, gc1, bec1, (long)N_GRAPHS * HID / 4, 0);
  gemm_wmma_kernel<HID, OUTF, 2><<<N_GRAPHS / 16, 32 * (OUTF / 16), 0, stream>>>(
      c1, Wc2, bc2, xcell);
}